// ExplicitRandomWalkEncoder_18202071400862
// MI455X (gfx1250) — compile-verified
//
#include <hip/hip_runtime.h>

typedef __attribute__((ext_vector_type(2))) float v2f;
typedef __attribute__((ext_vector_type(8))) float v8f;

#define HDIM 128
#define GDIM 384          // 3*H
#define TLEN 20
#define WPB  16           // walks per block
#define XPAD 4            // xg/h row stride 132 -> 16 A-rows hit distinct banks
#define GPAD 4            // gx/gh row stride 388 -> half-wave rows (+8) land +32 banks

// Async memory->LDS gather path (CDNA5 GLOBAL_LOAD_ASYNC_TO_LDS_B128, ASYNCcnt).
#if defined(__AMDGCN__) && defined(__gfx1250__) && \
    __has_builtin(__builtin_amdgcn_global_load_async_to_lds_b128) && \
    __has_builtin(__builtin_amdgcn_s_wait_asynccnt)
#define USE_ASYNC_GATHER 1
#else
#define USE_ASYNC_GATHER 0
#endif

#if USE_ASYNC_GATHER
// Match the builtin's parameter types exactly (from clang's diagnostic):
// pointer to int __attribute__((vector_size(16))) in AS1 (global) / AS3 (LDS).
typedef int v4i_vs __attribute__((vector_size(16)));
typedef __attribute__((address_space(1))) v4i_vs* async_gptr;
typedef __attribute__((address_space(3))) v4i_vs* async_lptr;
#endif

// ---------------------------------------------------------------------------
// Kernel 1: fold the feature encoder into the GRU input weights.
//   Wc = W_ih @ W_f            [384 x 128]
//   bc = W_ih @ b_f + b_ih     [384]
// Tiny one-off (12 MFLOP) -> simple VALU kernel. One block per output row.
// ---------------------------------------------------------------------------
__global__ void fuse_weights_kernel(const float* __restrict__ W_f,
                                    const float* __restrict__ b_f,
                                    const float* __restrict__ W_ih,
                                    const float* __restrict__ b_ih,
                                    float* __restrict__ Wc,
                                    float* __restrict__ bc) {
  __shared__ float red[HDIM];
  const int i = blockIdx.x;    // 0..383
  const int j = threadIdx.x;   // 0..127
  float acc = 0.0f;
  for (int k = 0; k < HDIM; ++k) {
    acc += W_ih[i * HDIM + k] * W_f[k * HDIM + j];
  }
  Wc[i * HDIM + j] = acc;
  red[j] = W_ih[i * HDIM + j] * b_f[j];
  __syncthreads();
  for (int s = 64; s > 0; s >>= 1) {
    if (j < s) red[j] += red[j + s];
    __syncthreads();
  }
  if (j == 0) bc[i] = red[0] + b_ih[i];
}

// ---------------------------------------------------------------------------
// Kernel 2: fused gather + 20-step GRU over 16 walks per block.
//  - All GEMMs on the f32 WMMA path (V_WMMA_F32_16X16X4_F32).
//  - Weights end up register-resident across the whole t-loop (compiler uses
//    the gfx1250 extended VGPR file via S_SET_VGPR_MSB).
//  - Gather of step t+1 is prefetched into a double-buffered LDS tile with
//    GLOBAL_LOAD_ASYNC_TO_LDS_B128 while step t's WMMA chains execute.
// ---------------------------------------------------------------------------
__global__ void __launch_bounds__(256)
walk_gru_kernel(const float* __restrict__ x,
                const int*   __restrict__ walks,
                const float* __restrict__ Wc,
                const float* __restrict__ bc,
                const float* __restrict__ W_hh,
                const float* __restrict__ b_hh,
                float* __restrict__ out) {
  __shared__ float xg[2][WPB][HDIM + XPAD]; // double-buffered gathered x rows
  __shared__ float hS[WPB][HDIM + XPAD];    // running hidden state
  __shared__ float gxs[WPB][GDIM + GPAD];   // staged input gates  (bias folded)
  __shared__ float ghs[WPB][GDIM + GPAD];   // staged hidden gates (bias folded)

  const int tid  = threadIdx.x;
  const int wave = tid >> 5;
  const int lane = tid & 31;
  const int w0   = blockIdx.x * WPB;        // first walk of this block

  // zero h
  for (int e = tid; e < WPB * (HDIM + XPAD); e += 256) {
    ((float*)hS)[e] = 0.0f;
  }

  // Gather one step's 16 x-rows into LDS buffer `buf`.
  // 512 float4 transfers, 2 per thread, contiguous within a row.
  auto gather = [&](int t, int buf) {
    for (int q = tid; q < WPB * (HDIM / 4); q += 256) {
      const int row = q >> 5;               // HDIM/4 = 32 float4 per row
      const int c4  = q & 31;
      const int node = walks[(w0 + row) * TLEN + t];
      const float* src = x + (size_t)node * HDIM + c4 * 4;
      float* dst = &xg[buf][row][c4 * 4];
#if USE_ASYNC_GATHER
      // Direct memory->LDS, tracked with ASYNCcnt; no VGPR staging.
      __builtin_amdgcn_global_load_async_to_lds_b128(
          (async_gptr)(uintptr_t)src,
          (async_lptr)(unsigned)(uintptr_t)dst,
          0, 0);
#else
      const float4 v = *(const float4*)src;
      dst[0] = v.x; dst[1] = v.y; dst[2] = v.z; dst[3] = v.w;
#endif
    }
  };

  // Prologue: start the gather for step 0.
  gather(0, 0);

  const int m  = lane & 15;                 // A-matrix row, B/D-matrix column
  const int kh = (lane >> 4) << 1;          // K sub-offset (0 or 2) per half-wave
  const int rbase = (lane >> 4) * 8;        // D-matrix row base for this half-wave

  for (int t = 0; t < TLEN; ++t) {
    const int buf = t & 1;

#if USE_ASYNC_GATHER
    __builtin_amdgcn_s_wait_asynccnt(0);    // this wave's async LDS writes done
#endif
    __syncthreads();                        // all waves' writes visible; h ready

    // Prefetch next step's rows into the other buffer while we compute.
    if (t + 1 < TLEN) gather(t + 1, buf ^ 1);

    // GEMMs: gx = xg @ Wc^T, gh = h @ W_hh^T.  M=16 walks, N=384, K=128.
    // 24 N-tiles, 3 per wave; K as 32 chained f32 WMMAs each.
    for (int s = 0; s < 3; ++s) {
      const int nbase = (wave * 3 + s) * 16;
      const int n = nbase + m;              // this lane's B-matrix column
      const float* wcrow = Wc   + (size_t)n * HDIM;
      const float* whrow = W_hh + (size_t)n * HDIM;
      v8f cx = {};
      v8f ch = {};
      for (int kk = 0; kk < HDIM; kk += 4) {
        const v2f a1 = *(const v2f*)&xg[buf][m][kk + kh];
        const v2f b1 = *(const v2f*)&wcrow[kk + kh];
        cx = __builtin_amdgcn_wmma_f32_16x16x4_f32(false, a1, false, b1,
                                                   (short)0, cx, false, false);
        const v2f a2 = *(const v2f*)&hS[m][kk + kh];
        const v2f b2 = *(const v2f*)&whrow[kk + kh];
        ch = __builtin_amdgcn_wmma_f32_16x16x4_f32(false, a2, false, b2,
                                                   (short)0, ch, false, false);
      }
      // D layout: VGPR i -> row i + 8*(lane>=16), col = n. Fold biases here.
      const float bcv = bc[n];
      const float bhv = b_hh[n];
      #pragma unroll
      for (int i = 0; i < 8; ++i) {
        gxs[rbase + i][n] = cx[i] + bcv;
        ghs[rbase + i][n] = ch[i] + bhv;
      }
    }
    __syncthreads();

    // GRU gates (torch order r,z,n): 2048 elements, 8 per thread.
    for (int e = tid; e < WPB * HDIM; e += 256) {
      const int mm = e >> 7;
      const int j  = e & 127;
      const float r  = 1.0f / (1.0f + __expf(-(gxs[mm][j]       + ghs[mm][j])));
      const float z  = 1.0f / (1.0f + __expf(-(gxs[mm][j + 128] + ghs[mm][j + 128])));
      const float nn = tanhf(gxs[mm][j + 256] + r * ghs[mm][j + 256]);
      hS[mm][j] = (1.0f - z) * nn + z * hS[mm][j];
    }
  }
  __syncthreads();

  // Emit final hidden states.
  for (int e = tid; e < WPB * HDIM; e += 256) {
    const int mm = e >> 7;
    const int j  = e & 127;
    out[(size_t)(w0 + mm) * HDIM + j] = hS[mm][j];
  }
}

// ---------------------------------------------------------------------------
extern "C" void kernel_launch(void* const* d_in, const int* in_sizes, int n_in,
                              void* d_out, int out_size, void* d_ws, size_t ws_size,
                              hipStream_t stream) {
  const float* x     = (const float*)d_in[0];
  const int*   walks = (const int*)  d_in[1];
  const float* W_f   = (const float*)d_in[2];
  const float* b_f   = (const float*)d_in[3];
  const float* W_ih  = (const float*)d_in[4];
  const float* W_hh  = (const float*)d_in[5];
  const float* b_ih  = (const float*)d_in[6];
  const float* b_hh  = (const float*)d_in[7];
  float* out = (float*)d_out;

  float* Wc = (float*)d_ws;                 // 384*128 floats
  float* bc = Wc + GDIM * HDIM;             // 384 floats

  fuse_weights_kernel<<<GDIM, HDIM, 0, stream>>>(W_f, b_f, W_ih, b_ih, Wc, bc);

  const int num_walks = in_sizes[1] / TLEN; // 16384
  const int nblocks = num_walks / WPB;      // 1024
  walk_gru_kernel<<<nblocks, 256, 0, stream>>>(x, walks, Wc, bc, W_hh, b_hh, out);
}